// ContinuousLS_45956150068053
// MI455X (gfx1250) — compile-verified
//
#include <hip/hip_runtime.h>
#include <math.h>

// ---------------------------------------------------------------------------
// ContinuousLS pipeline for MI455X (gfx1250, wave32).
// Heavy ops (K = A^T A, K2 = K K) run on V_WMMA_F32_16X16X4_F32 with the
// k-panel staged into LDS via CDNA5 async global->LDS copies (ASYNCcnt).
// All reductions are fixed-order (deterministic). Threefry-2x32 replicates
// JAX's PRNG for the Gumbel draws and the final randint.
// Shapes fixed by the reference: n=8192, d=1024, s=8, num_samples=80.
// ---------------------------------------------------------------------------

#define N_ROWS 8192
#define D_COLS 1024
#define S_COLS 8
#define NSAMP  80
#define KT     16   // k-depth staged in LDS per stage (4 WMMA steps)

typedef __attribute__((ext_vector_type(2))) float v2f;
typedef __attribute__((ext_vector_type(8))) float v8f;

// ------------------------------ Threefry-2x32 ------------------------------
__device__ __forceinline__ void tf2x32(unsigned k0, unsigned k1,
                                       unsigned x0, unsigned x1,
                                       unsigned* o0, unsigned* o1) {
  unsigned ks0 = k0, ks1 = k1, ks2 = k0 ^ k1 ^ 0x1BD11BDAu;
  const int ra[4] = {13, 15, 26, 6};
  const int rb[4] = {17, 29, 16, 24};
  x0 += ks0; x1 += ks1;
#define TF_RND(r) { x0 += x1; x1 = (x1 << (r)) | (x1 >> (32 - (r))); x1 ^= x0; }
  for (int i = 0; i < 4; ++i) TF_RND(ra[i]);
  x0 += ks1; x1 += ks2 + 1u;
  for (int i = 0; i < 4; ++i) TF_RND(rb[i]);
  x0 += ks2; x1 += ks0 + 2u;
  for (int i = 0; i < 4; ++i) TF_RND(ra[i]);
  x0 += ks0; x1 += ks1 + 3u;
  for (int i = 0; i < 4; ++i) TF_RND(rb[i]);
  x0 += ks1; x1 += ks2 + 4u;
  for (int i = 0; i < 4; ++i) TF_RND(ra[i]);
  x0 += ks2; x1 += ks0 + 5u;
#undef TF_RND
  *o0 = x0; *o1 = x1;
}

// jax.random.key(42) -> (0,42); split -> kg = (a0,b0), km = (a1,b1)
__device__ __forceinline__ void derive_keys(unsigned* kg0, unsigned* kg1,
                                            unsigned* km0, unsigned* km1) {
  unsigned a0, a1, b0, b1;
  tf2x32(0u, 42u, 0u, 2u, &a0, &a1);
  tf2x32(0u, 42u, 1u, 3u, &b0, &b1);
  *kg0 = a0; *kg1 = b0; *km0 = a1; *km1 = b1;
}

// -------------------- Jacobi eigensolver (m<=8, symmetric) -----------------
__device__ void jacobi_sym(float* A, float* V, int m) {
  for (int i = 0; i < m; ++i)
    for (int j = 0; j < m; ++j) V[i * m + j] = (i == j) ? 1.0f : 0.0f;
  for (int sweep = 0; sweep < 14; ++sweep) {
    float off = 0.0f;
    for (int p = 0; p < m; ++p)
      for (int q = p + 1; q < m; ++q) { float x = A[p * m + q]; off += x * x; }
    float dia = 0.0f;
    for (int p = 0; p < m; ++p) { float x = A[p * m + p]; dia += x * x; }
    if (off <= 1e-14f * (dia + 1e-30f)) break;
    for (int p = 0; p < m - 1; ++p) {
      for (int q = p + 1; q < m; ++q) {
        float apq = A[p * m + q];
        float app = A[p * m + p], aqq = A[q * m + q];
        if (fabsf(apq) <= 1e-9f * (fabsf(app) + fabsf(aqq) + 1e-30f)) continue;
        float tau = (aqq - app) / (2.0f * apq);
        float t = (tau >= 0.0f ? 1.0f : -1.0f) /
                  (fabsf(tau) + sqrtf(1.0f + tau * tau));
        float c = 1.0f / sqrtf(1.0f + t * t);
        float s = t * c;
        for (int k = 0; k < m; ++k) {
          float akp = A[k * m + p], akq = A[k * m + q];
          A[k * m + p] = c * akp - s * akq;
          A[k * m + q] = s * akp + c * akq;
        }
        for (int k = 0; k < m; ++k) {
          float apk = A[p * m + k], aqk = A[q * m + k];
          A[p * m + k] = c * apk - s * aqk;
          A[q * m + k] = s * apk + c * aqk;
        }
        for (int k = 0; k < m; ++k) {
          float vkp = V[k * m + p], vkq = V[k * m + q];
          V[k * m + p] = c * vkp - s * vkq;
          V[k * m + q] = s * vkp + c * vkq;
        }
      }
    }
  }
}

// ------------------------ async copy helper (CDNA5) ------------------------
// One element: copies 4 bytes from the global address gp to the LDS address lds_addr; tracked by ASYNCcnt.
__device__ __forceinline__ void async_g2l_b32(unsigned lds_addr,
                                              const float* gp) {
  unsigned long long ga = (unsigned long long)(size_t)gp;
  asm volatile("global_load_async_to_lds_b32 %0, %1, off"
               :: "v"(lds_addr), "v"(ga) : "memory");
}
__device__ __forceinline__ void wait_async0() {
  asm volatile("s_wait_asynccnt 0x0" ::: "memory");
}

// --------------------------- WMMA SYRK: Y = X^T X --------------------------
// X is n x d (row-major). Output Y is d x d. 256 threads = 8 waves per block;
// block computes a 32(I) x 64(J) macro-tile, one 16x16 WMMA tile per wave.
// Per KT=16 stage the block async-copies the k-panel (32 I-cols + 64 J-cols,
// 6 KB) into LDS *transposed* ([col][k]) so each lane's two k-values for the
// 16x16x4 f32 fragment are one ds_load_b64. Waves share the panel -> 8x less
// global traffic; A' (33.5 MB) then lives in MI455X's 192 MB L2.
__global__ __launch_bounds__(256) void k_syrk_wmma(const float* __restrict__ X,
                                                   float* __restrict__ Y,
                                                   int n, int d) {
  __shared__ float shA[32 * KT];   // shA[c*KT + k]
  __shared__ float shB[64 * KT];   // shB[c*KT + k]
  int tid  = threadIdx.x;
  int w    = tid >> 5;             // wave 0..7
  int lane = tid & 31;
  int I0 = blockIdx.x * 32;
  int J0 = blockIdx.y * 64;
  int m  = lane & 15;
  int kh = (lane >> 4) << 1;       // 0 or 2 (A/B fragment K-half per lane)
  int ca = (w & 1) * 16 + m;       // this lane's column inside shA
  int cb = (w >> 1) * 16 + m;      // this lane's column inside shB
  v8f acc = {};
  for (int k0 = 0; k0 < n; k0 += KT) {
    // ---- stage k-panel into LDS (transposed) via async global->LDS ----
#pragma unroll
    for (int it = 0; it < 2; ++it) {          // 512 A-window elements
      int e = tid + it * 256;
      int r = e >> 5, c = e & 31;             // row k0+r, col I0+c
      async_g2l_b32((unsigned)(size_t)&shA[c * KT + r],
                    X + (size_t)(k0 + r) * d + I0 + c);
    }
#pragma unroll
    for (int it = 0; it < 4; ++it) {          // 1024 B-window elements
      int e = tid + it * 256;
      int r = e >> 6, c = e & 63;             // row k0+r, col J0+c
      async_g2l_b32((unsigned)(size_t)&shB[c * KT + r],
                    X + (size_t)(k0 + r) * d + J0 + c);
    }
    wait_async0();
    __syncthreads();
    // ---- 4 WMMA steps over the staged panel ----
#pragma unroll
    for (int kk = 0; kk < KT; kk += 4) {
      v2f a = *(const v2f*)&shA[ca * KT + kk + kh];
      v2f b = *(const v2f*)&shB[cb * KT + kk + kh];
      acc = __builtin_amdgcn_wmma_f32_16x16x4_f32(
          false, a, false, b, (short)0, acc, false, false);
    }
    __syncthreads();                          // panel consumed; safe to refill
  }
  int row0 = I0 + (w & 1) * 16 + ((lane >> 4) << 3);
  int col  = J0 + (w >> 1) * 16 + m;
#pragma unroll
  for (int r = 0; r < 8; ++r)
    Y[(size_t)(row0 + r) * d + col] = acc[r];
}

// ------------------- column squared norms: out[c] = sum X[:,c]^2 -----------
// Shared by A' (ld=1024) and S (ld=8) so copied columns reduce in the exact
// same operation order -> bitwise-equal sums -> the 1e-5 match always fires.
__global__ __launch_bounds__(256) void k_colsq(const float* __restrict__ X,
                                               int ld, int n,
                                               float* __restrict__ out) {
  __shared__ float red[256];
  int col = blockIdx.x, t = threadIdx.x;
  float acc = 0.0f;
  for (int i = t; i < n; i += 256) {
    float v = X[(size_t)i * ld + col];
    acc += v * v;
  }
  red[t] = acc; __syncthreads();
  for (int s = 128; s > 0; s >>= 1) {
    if (t < s) red[t] += red[t + s];
    __syncthreads();
  }
  if (t == 0) out[col] = red[0];
}

// ----------------------------- G_S = S^T S (8x8) ---------------------------
__global__ __launch_bounds__(256) void k_gramS(const float* __restrict__ S,
                                               float* __restrict__ G) {
  __shared__ float red[256];
  int r = blockIdx.x >> 3, c = blockIdx.x & 7, t = threadIdx.x;
  float acc = 0.0f;
  for (int i = t; i < N_ROWS; i += 256)
    acc += S[(size_t)i * S_COLS + r] * S[(size_t)i * S_COLS + c];
  red[t] = acc; __syncthreads();
  for (int s = 128; s > 0; s >>= 1) {
    if (t < s) red[t] += red[t + s];
    __syncthreads();
  }
  if (t == 0) G[blockIdx.x] = red[0];
}

// -------------------------- StA = S^T A' (8 x 1024) ------------------------
__global__ __launch_bounds__(256) void k_stA(const float* __restrict__ A,
                                             const float* __restrict__ S,
                                             float* __restrict__ StA) {
  __shared__ float red[256];
  int j = blockIdx.x, t = threadIdx.x;
  float acc[S_COLS] = {};
  for (int i = t; i < N_ROWS; i += 256) {
    float av = A[(size_t)i * D_COLS + j];
#pragma unroll
    for (int r = 0; r < S_COLS; ++r) acc[r] += S[(size_t)i * S_COLS + r] * av;
  }
  for (int r = 0; r < S_COLS; ++r) {
    red[t] = acc[r]; __syncthreads();
    for (int s = 128; s > 0; s >>= 1) {
      if (t < s) red[t] += red[t + s];
      __syncthreads();
    }
    if (t == 0) StA[r * D_COLS + j] = red[0];
    __syncthreads();
  }
}

// -------------------- Ginv = pinv(G_S) via Jacobi eigen --------------------
__global__ void k_pinv8(const float* __restrict__ G_S,
                        float* __restrict__ Ginv) {
  float A[64], V[64];
  for (int i = 0; i < 64; ++i) A[i] = G_S[i];
  jacobi_sym(A, V, 8);
  float lmax = 0.0f;
  for (int e = 0; e < 8; ++e) lmax = fmaxf(lmax, fabsf(A[e * 8 + e]));
  float cutoff = 10.0f * 8.0f * 1.1920929e-7f * lmax;  // jax pinv rcond
  for (int i = 0; i < 8; ++i)
    for (int j = 0; j < 8; ++j) {
      float s = 0.0f;
      for (int e = 0; e < 8; ++e) {
        float lam = A[e * 8 + e];
        if (fabsf(lam) > cutoff) s += V[i * 8 + e] * V[j * 8 + e] / lam;
      }
      Ginv[i * 8 + j] = s;
    }
}

// ----------------------- P = Ginv * StA (8 x 1024) -------------------------
__global__ __launch_bounds__(256) void k_proj(const float* __restrict__ Ginv,
                                              const float* __restrict__ StA,
                                              float* __restrict__ P) {
  int tid = blockIdx.x * 256 + threadIdx.x;
  if (tid >= S_COLS * D_COLS) return;
  int r = tid >> 10, j = tid & (D_COLS - 1);
  float s = 0.0f;
  for (int c = 0; c < S_COLS; ++c) s += Ginv[r * 8 + c] * StA[c * D_COLS + j];
  P[tid] = s;
}

// ----------- cn[j] = sum_i (A[i][j] - sum_r S[i][r] P[r][j])^2 -------------
__global__ __launch_bounds__(256) void k_resid(const float* __restrict__ A,
                                               const float* __restrict__ S,
                                               const float* __restrict__ P,
                                               float* __restrict__ cn) {
  __shared__ float red[256];
  int j = blockIdx.x, t = threadIdx.x;
  float p[S_COLS];
#pragma unroll
  for (int r = 0; r < S_COLS; ++r) p[r] = P[r * D_COLS + j];
  float acc = 0.0f;
  for (int i = t; i < N_ROWS; i += 256) {
    float e = A[(size_t)i * D_COLS + j];
#pragma unroll
    for (int r = 0; r < S_COLS; ++r) e -= S[(size_t)i * S_COLS + r] * p[r];
    acc += e * e;
  }
  red[t] = acc; __syncthreads();
  for (int s = 128; s > 0; s >>= 1) {
    if (t < s) red[t] += red[t + s];
    __syncthreads();
  }
  if (t == 0) cn[j] = red[0];
}

// ----------------- deterministic sums: scal[0]=sum cn, [1]=A_fro2 ----------
__global__ __launch_bounds__(256) void k_sums(const float* __restrict__ cn,
                                              const float* __restrict__ a_sq,
                                              float* __restrict__ scal) {
  __shared__ float red[256];
  int t = threadIdx.x;
  float a = cn[t] + cn[t + 256] + cn[t + 512] + cn[t + 768];
  red[t] = a; __syncthreads();
  for (int s = 128; s > 0; s >>= 1) {
    if (t < s) red[t] += red[t + s];
    __syncthreads();
  }
  if (t == 0) scal[0] = red[0];
  __syncthreads();
  float b = a_sq[t] + a_sq[t + 256] + a_sq[t + 512] + a_sq[t + 768];
  red[t] = b; __syncthreads();
  for (int s = 128; s > 0; s >>= 1) {
    if (t < s) red[t] += red[t + s];
    __syncthreads();
  }
  if (t == 0) scal[1] = red[0];
}

// ------------- logits[j] = log(probs[j]+eps) + gumbel_j (JAX RNG) ----------
__global__ __launch_bounds__(256) void k_logits(const float* __restrict__ cn,
                                                const float* __restrict__ scal,
                                                float* __restrict__ logits) {
  int j = blockIdx.x * 256 + threadIdx.x;
  if (j >= D_COLS) return;
  unsigned kg0, kg1, km0, km1;
  derive_keys(&kg0, &kg1, &km0, &km1);
  // jax threefry bits for shape (1024,): pair (i, i+512)
  unsigned i = (j < 512) ? (unsigned)j : (unsigned)(j - 512);
  unsigned u0, u1;
  tf2x32(kg0, kg1, i, i + 512u, &u0, &u1);
  unsigned bits = (j < 512) ? u0 : u1;
  unsigned fb = (bits >> 9) | 0x3f800000u;
  float u = __uint_as_float(fb) - 1.0f;          // [0,1)
  float g = -logf(-logf(u + 1e-10f) + 1e-10f);   // gumbel
  float prob = cn[j] / (scal[0] + 1e-10f);
  logits[j] = logf(prob + 1e-10f) + g;
}

// ---------------- top-80 of logits, stable (tie -> lower index) ------------
__global__ __launch_bounds__(256) void k_top80(const float* __restrict__ logits,
                                               int* __restrict__ C_idx) {
  __shared__ float lv[D_COLS];
  __shared__ float sv[256];
  __shared__ int   si[256];
  int t = threadIdx.x;
  for (int j = t; j < D_COLS; j += 256) lv[j] = logits[j];
  __syncthreads();
  for (int it = 0; it < NSAMP; ++it) {
    float bv = -INFINITY; int bi = D_COLS;
    for (int j = t; j < D_COLS; j += 256) {
      float v = lv[j];
      if (v > bv || (v == bv && j < bi)) { bv = v; bi = j; }
    }
    sv[t] = bv; si[t] = bi;
    __syncthreads();
    for (int s = 128; s > 0; s >>= 1) {
      if (t < s) {
        float v2 = sv[t + s]; int i2 = si[t + s];
        if (v2 > sv[t] || (v2 == sv[t] && i2 < si[t])) { sv[t] = v2; si[t] = i2; }
      }
      __syncthreads();
    }
    if (t == 0) { C_idx[it] = si[0]; lv[si[0]] = -INFINITY; }
    __syncthreads();
  }
}

// ----------------- I_soft flags + sel_idx (top-8 of flags, sorted) ---------
__device__ void pick8_sorted(const int* flag1024, int skip_zero_idx,
                             int force_one_idx, int* out8) {
  int cnt = 0;
  for (int j = 0; j < D_COLS && cnt < 8; ++j) {
    int f = flag1024[j];
    if (j == skip_zero_idx) f = 0;
    if (j == force_one_idx) f = 1;
    if (f) out8[cnt++] = j;
  }
  for (int j = 0; j < D_COLS && cnt < 8; ++j) {
    int f = flag1024[j];
    if (j == skip_zero_idx) f = 0;
    if (j == force_one_idx) f = 1;
    if (!f) out8[cnt++] = j;
  }
  for (int i = 1; i < 8; ++i) {   // insertion sort
    int v = out8[i], p = i - 1;
    while (p >= 0 && out8[p] > v) { out8[p + 1] = out8[p]; --p; }
    out8[p + 1] = v;
  }
}

__global__ __launch_bounds__(256) void k_match(const float* __restrict__ a_sq,
                                               const float* __restrict__ s_sq,
                                               int* __restrict__ isoft,
                                               int* __restrict__ sel) {
  int t = threadIdx.x;
  for (int j = t; j < D_COLS; j += 256) {
    float an = sqrtf(a_sq[j]);
    int f = 0;
    for (int r = 0; r < S_COLS; ++r) {
      float sn = sqrtf(s_sq[r]);
      if (fabsf(sn - an) / (an + 1e-10f) < 1e-5f) f = 1;
    }
    isoft[j] = f;
  }
  __syncthreads();
  if (t == 0) {
    int tmp[8];
    pick8_sorted(isoft, -1, -1, tmp);
    for (int i = 0; i < 8; ++i) sel[i] = tmp[i];
  }
}

// --------------- 640 masked pinv objectives (one per thread) ---------------
__global__ __launch_bounds__(64) void k_objs(const float* __restrict__ K,
                                             const float* __restrict__ K2,
                                             const int* __restrict__ sel,
                                             const int* __restrict__ C_idx,
                                             const float* __restrict__ scal,
                                             float* __restrict__ objs) {
  int t = blockIdx.x * 64 + threadIdx.x;
  if (t >= NSAMP * S_COLS) return;
  int p_i = t / S_COLS, qpos = t % S_COLS;   // row-major (p, q) like reference
  int p = C_idx[p_i];
  int q_idx = sel[qpos];
  int act[9]; int m = 0;
  for (int r = 0; r < S_COLS; ++r)
    if (r != qpos) act[m++] = sel[r];
  if (p != q_idx) act[m++] = p;              // slot s masked iff p == q_idx
  float G[64], M[64], V[64];
  for (int a = 0; a < m; ++a)
    for (int b = 0; b < m; ++b) {
      G[a * m + b] = K [(size_t)act[a] * D_COLS + act[b]];
      M[a * m + b] = K2[(size_t)act[a] * D_COLS + act[b]];
    }
  jacobi_sym(G, V, m);
  float lmax = 0.0f;
  for (int e = 0; e < m; ++e) lmax = fmaxf(lmax, fabsf(G[e * m + e]));
  float cutoff = 10.0f * 9.0f * 1.1920929e-7f * lmax;  // full matrix is 9x9
  float s = 0.0f;
  for (int e = 0; e < m; ++e) {
    float lam = G[e * m + e];
    if (fabsf(lam) > cutoff) {
      float quad = 0.0f;
      for (int a = 0; a < m; ++a) {
        float row = 0.0f;
        for (int b = 0; b < m; ++b) row += M[a * m + b] * V[b * m + e];
        quad += V[a * m + e] * row;
      }
      s += quad / lam;
    }
  }
  float val = scal[1] - s;
  objs[t] = sqrtf(fmaxf(val, 0.0f));
}

// ---------------- argmin, randint, final index set (1 thread) --------------
__global__ void k_final(const float* __restrict__ objs,
                        const int* __restrict__ sel,
                        const int* __restrict__ C_idx,
                        const int* __restrict__ isoft,
                        int* __restrict__ outidx) {
  float best = objs[0]; int amin = 0;
  for (int i = 1; i < NSAMP * S_COLS; ++i)
    if (objs[i] < best) { best = objs[i]; amin = i; }
  int min_idx = sel[amin % S_COLS];
  // jax.random.randint(km, (), 0, 80): split km -> k1,k2; bits at count 0;
  // multiplier = (2^16 % 80)^2 % 80 = 16
  unsigned kg0, kg1, km0, km1;
  derive_keys(&kg0, &kg1, &km0, &km1);
  unsigned A0, A1, B0, B1, h, l, junk;
  tf2x32(km0, km1, 0u, 2u, &A0, &A1);
  tf2x32(km0, km1, 1u, 3u, &B0, &B1);
  tf2x32(A0, B0, 0u, 0u, &h, &junk);   // k1 -> higher bits
  tf2x32(A1, B1, 0u, 0u, &l, &junk);   // k2 -> lower bits
  unsigned off = ((h % 80u) * 16u + (l % 80u)) % 80u;
  int best_p = C_idx[off];
  int tmp[8];
  pick8_sorted(isoft, min_idx, best_p, tmp);
  for (int i = 0; i < 8; ++i) outidx[i] = tmp[i];
}

// ------------------------- gather: out = A'[:, outidx] ---------------------
__global__ __launch_bounds__(256) void k_gather(const float* __restrict__ A,
                                                const int* __restrict__ outidx,
                                                float* __restrict__ out) {
  int gid = blockIdx.x * 256 + threadIdx.x;
  if (gid >= N_ROWS * S_COLS) return;
  int i = gid >> 3, c = gid & 7;
  out[gid] = A[(size_t)i * D_COLS + outidx[c]];
}

// ---------------------------------------------------------------------------
extern "C" void kernel_launch(void* const* d_in, const int* in_sizes, int n_in,
                              void* d_out, int out_size, void* d_ws,
                              size_t ws_size, hipStream_t stream) {
  (void)in_sizes; (void)n_in; (void)out_size; (void)ws_size;
  const float* A = (const float*)d_in[0];   // 8192 x 1024 row-major
  const float* S = (const float*)d_in[2];   // 8192 x 8 row-major
  float* out = (float*)d_out;               // 8192 x 8 row-major

  // workspace layout (floats)
  float* ws     = (float*)d_ws;
  float* K      = ws;                       // 1024*1024
  float* K2     = K + (1 << 20);            // 1024*1024
  float* StA    = K2 + (1 << 20);           // 8*1024
  float* P      = StA + 8 * 1024;           // 8*1024
  float* a_sq   = P + 8 * 1024;             // 1024
  float* s_sq   = a_sq + 1024;              // 8
  float* G_S    = s_sq + 8;                 // 64
  float* Ginv   = G_S + 64;                 // 64
  float* cn     = Ginv + 64;                // 1024
  float* logits = cn + 1024;                // 1024
  float* scal   = logits + 1024;            // 4
  float* objs   = scal + 4;                 // 640
  int*   C_idx  = (int*)(objs + 640);       // 80
  int*   sel    = C_idx + 80;               // 8
  int*   isoft  = sel + 8;                  // 1024
  int*   outidx = isoft + 1024;             // 8

  dim3 gK(32, 16);  // 8 waves/block -> 32x64 macro tile; covers 1024x1024
  k_syrk_wmma<<<gK, 256, 0, stream>>>(A, K, N_ROWS, D_COLS);   // K  = A^T A
  k_syrk_wmma<<<gK, 256, 0, stream>>>(K, K2, D_COLS, D_COLS);  // K2 = K^T K = K K

  k_colsq<<<D_COLS, 256, 0, stream>>>(A, D_COLS, N_ROWS, a_sq);
  k_colsq<<<S_COLS, 256, 0, stream>>>(S, S_COLS, N_ROWS, s_sq);
  k_gramS<<<64, 256, 0, stream>>>(S, G_S);
  k_stA<<<D_COLS, 256, 0, stream>>>(A, S, StA);
  k_pinv8<<<1, 1, 0, stream>>>(G_S, Ginv);
  k_proj<<<(S_COLS * D_COLS + 255) / 256, 256, 0, stream>>>(Ginv, StA, P);
  k_resid<<<D_COLS, 256, 0, stream>>>(A, S, P, cn);
  k_sums<<<1, 256, 0, stream>>>(cn, a_sq, scal);
  k_logits<<<(D_COLS + 255) / 256, 256, 0, stream>>>(cn, scal, logits);
  k_top80<<<1, 256, 0, stream>>>(logits, C_idx);
  k_match<<<1, 256, 0, stream>>>(a_sq, s_sq, isoft, sel);
  k_objs<<<(NSAMP * S_COLS + 63) / 64, 64, 0, stream>>>(K, K2, sel, C_idx,
                                                        scal, objs);
  k_final<<<1, 1, 0, stream>>>(objs, sel, C_idx, isoft, outidx);
  k_gather<<<(N_ROWS * S_COLS + 255) / 256, 256, 0, stream>>>(A, outidx, out);
}